// Graph_UNet_4037269258361
// MI455X (gfx1250) — compile-verified
//
#include <hip/hip_runtime.h>
#include <hip/hip_bf16.h>
#include <math.h>

// ---------------------------------------------------------------------------
// CDNA5 WMMA f32 16x16x4:  A = 16x4 (2 VGPR), B = 4x16 (2 VGPR),
// C/D = 16x16 f32 (8 VGPR).  Wave32 only.
// ---------------------------------------------------------------------------
typedef float v2f __attribute__((ext_vector_type(2)));
typedef float v8f __attribute__((ext_vector_type(8)));

// ---------------------------------------------------------------------------
// Tiled f32 WMMA GEMM:  C[MxN] = A[MxK] * B'[KxN]
//   B'(k,n) = transB ? B[n*ldb + k] : B[k*ldb + n]
// Block: 256 threads = 8 waves, macro-tile TM(64) x TN_ (64/128/256), TK=32.
// Waves arranged 2 x 4; each wave computes 32 x (TN_/4) with 2 x JT
// independent 16x16 accumulators (JT = TN_/64).
// Per K-step of 4: 2 A-frags + JT B-frags feed 2*JT WMMAs.
// Both LDS tiles are stored k-contiguous with +1 padding so every fragment
// is one merged ds_load_2addr_b32 and strides are conflict-free.
// ---------------------------------------------------------------------------
template <int TN_>
__global__ __launch_bounds__(256)
void wmma_gemm_f32(const float* __restrict__ A, const float* __restrict__ B,
                   float* __restrict__ C, int M, int N, int K,
                   int lda, int ldb, int ldc, int transB) {
    constexpr int TM_ = 64;
    constexpr int TK_ = 32;
    constexpr int JT  = TN_ / 64;      // 16-col tiles per wave

    __shared__ float As [TM_][TK_ + 1];   // [row][k]
    __shared__ float BsT[TN_][TK_ + 1];   // [col][k]  (k-contiguous)

    const int tid  = threadIdx.x;
    const int wave = tid >> 5;
    const int lane = tid & 31;
    const int half = lane >> 4;        // 0: lanes 0-15, 1: lanes 16-31
    const int l16  = lane & 15;
    const int wm   = wave >> 2;        // 0..1  (32-row slab)
    const int wn   = wave & 3;         // 0..3  (TN_/4-col slab)

    const int blockRow = blockIdx.y * TM_;
    const int blockCol = blockIdx.x * TN_;

    v8f acc[2][JT];
    #pragma unroll
    for (int i = 0; i < 2; ++i)
        #pragma unroll
        for (int j = 0; j < JT; ++j) { v8f z = {}; acc[i][j] = z; }

    const bool fullMN = (blockRow + TM_ <= M) && (blockCol + TN_ <= N) &&
                        !transB && ((lda & 3) == 0) && ((ldb & 3) == 0);

    for (int k0 = 0; k0 < K; k0 += TK_) {
        if (fullMN && (k0 + TK_ <= K)) {
            // ---- fast path: aligned float4 global loads -------------------
            #pragma unroll
            for (int p = 0; p < 2; ++p) {              // A: 64x32 = 512 f4
                int e  = tid + p * 256;
                int r  = e >> 3;
                int c4 = (e & 7) << 2;
                const float4 v = *reinterpret_cast<const float4*>(
                    A + (size_t)(blockRow + r) * lda + k0 + c4);
                As[r][c4 + 0] = v.x; As[r][c4 + 1] = v.y;
                As[r][c4 + 2] = v.z; As[r][c4 + 3] = v.w;
            }
            #pragma unroll
            for (int p = 0; p < TN_ / 32; ++p) {       // B: 32 x TN_ = 8*TN_ f4
                int e  = tid + p * 256;
                int kk = e / (TN_ / 4);
                int c4 = (e % (TN_ / 4)) << 2;
                const float4 v = *reinterpret_cast<const float4*>(
                    B + (size_t)(k0 + kk) * ldb + blockCol + c4);
                BsT[c4 + 0][kk] = v.x; BsT[c4 + 1][kk] = v.y;
                BsT[c4 + 2][kk] = v.z; BsT[c4 + 3][kk] = v.w;
            }
        } else {
            // ---- generic guarded path ------------------------------------
            for (int e = tid; e < TM_ * TK_; e += 256) {
                int r = e >> 5, c = e & 31;
                int gr = blockRow + r, gc = k0 + c;
                As[r][c] = (gr < M && gc < K) ? A[(size_t)gr * lda + gc] : 0.0f;
            }
            if (transB) {
                // coalesced along k for row-major W (x @ W^T)
                for (int e = tid; e < TK_ * TN_; e += 256) {
                    int c = e >> 5, kk = e & 31;
                    int gk = k0 + kk, gn = blockCol + c;
                    BsT[c][kk] = (gk < K && gn < N) ? B[(size_t)gn * ldb + gk] : 0.0f;
                }
            } else {
                for (int e = tid; e < TK_ * TN_; e += 256) {
                    int kk = e / TN_, c = e % TN_;
                    int gk = k0 + kk, gn = blockCol + c;
                    BsT[c][kk] = (gk < K && gn < N) ? B[(size_t)gk * ldb + gn] : 0.0f;
                }
            }
        }
        __syncthreads();

        #pragma unroll
        for (int kk = 0; kk < TK_; kk += 4) {
            v2f af[2], bf[JT];
            #pragma unroll
            for (int i = 0; i < 2; ++i) {
                int r = wm * 32 + i * 16 + l16;
                af[i].x = As[r][kk + 2 * half + 0];
                af[i].y = As[r][kk + 2 * half + 1];
            }
            #pragma unroll
            for (int j = 0; j < JT; ++j) {
                int c = wn * (TN_ / 4) + j * 16 + l16;
                bf[j].x = BsT[c][kk + 2 * half + 0];
                bf[j].y = BsT[c][kk + 2 * half + 1];
            }
            #pragma unroll
            for (int i = 0; i < 2; ++i)
                #pragma unroll
                for (int j = 0; j < JT; ++j)
                    acc[i][j] = __builtin_amdgcn_wmma_f32_16x16x4_f32(
                        false, af[i], false, bf[j], (short)0, acc[i][j], false, false);
        }
        __syncthreads();
    }

    // Store: VGPR v of acc holds row (v + half*8), col = l16 of its 16x16 tile
    #pragma unroll
    for (int i = 0; i < 2; ++i) {
        #pragma unroll
        for (int j = 0; j < JT; ++j) {
            int col = blockCol + wn * (TN_ / 4) + j * 16 + l16;
            if (col < N) {
                #pragma unroll
                for (int v = 0; v < 8; ++v) {
                    int row = blockRow + wm * 32 + i * 16 + half * 8 + v;
                    if (row < M) C[(size_t)row * ldc + col] = acc[i][j][v];
                }
            }
        }
    }
}

// ---------------------------------------------------------------------------
// Small utility kernels
// ---------------------------------------------------------------------------
__global__ void fill_zero_kernel(float* p, int total) {
    int i = blockIdx.x * 256 + threadIdx.x;
    if (i < total) p[i] = 0.0f;
}

__global__ void build_adj_kernel(const int* __restrict__ ei, int E, float* A, int n) {
    int e = blockIdx.x * 256 + threadIdx.x;
    if (e < E) {
        int r = ei[e];          // edge_index[0][e]
        int c = ei[E + e];      // edge_index[1][e]
        atomicAdd(&A[(size_t)r * n + c], 1.0f);   // exact: integer counts in f32
    }
}

__global__ void rowsum_dis_kernel(const float* __restrict__ A, int n, float* dis) {
    __shared__ float red[256];
    int row = blockIdx.x;
    float s = 0.0f;
    for (int j = threadIdx.x; j < n; j += 256) s += A[(size_t)row * n + j];
    red[threadIdx.x] = s;
    __syncthreads();
    for (int off = 128; off > 0; off >>= 1) {
        if (threadIdx.x < off) red[threadIdx.x] += red[threadIdx.x + off];
        __syncthreads();
    }
    if (threadIdx.x == 0) {
        float d = red[0] + 2.0f;                 // A_hat = A + 2I
        dis[row] = (d > 0.0f) ? rsqrtf(d) : 0.0f;
    }
}

__global__ void scale_rows_kernel(const float* __restrict__ X, const float* __restrict__ dis,
                                  float* Z, int total, int F) {
    int idx = blockIdx.x * 256 + threadIdx.x;
    if (idx < total) Z[idx] = X[idx] * dis[idx / F];
}

__global__ void gcn_epilogue_kernel(const float* __restrict__ Y, const float* __restrict__ XW,
                                    const float* __restrict__ dis, const float* __restrict__ bias,
                                    float* out, int total, int F, int do_relu) {
    int idx = blockIdx.x * 256 + threadIdx.x;
    if (idx < total) {
        int i = idx / F, f = idx % F;
        float di = dis[i];
        float v = di * (Y[idx] + 2.0f * di * XW[idx]) + bias[f];
        out[idx] = do_relu ? fmaxf(v, 0.0f) : v;
    }
}

__global__ void make_a1_kernel(const float* __restrict__ A, float* A1, int n, int total) {
    int idx = blockIdx.x * 256 + threadIdx.x;
    if (idx < total) {
        int i = idx / n, j = idx % n;
        A1[idx] = (i == j) ? 1.0f : A[idx];
    }
}

__global__ void zero_diag_kernel(float* A, int n) {
    int i = blockIdx.x * 256 + threadIdx.x;
    if (i < n) A[(size_t)i * n + i] = 0.0f;
}

__global__ void invnorm_kernel(const float* __restrict__ w, int F, float* out) {
    __shared__ float red[256];
    float s = 0.0f;
    for (int i = threadIdx.x; i < F; i += 256) s += w[i] * w[i];
    red[threadIdx.x] = s;
    __syncthreads();
    for (int off = 128; off > 0; off >>= 1) {
        if (threadIdx.x < off) red[threadIdx.x] += red[threadIdx.x + off];
        __syncthreads();
    }
    if (threadIdx.x == 0) out[0] = rsqrtf(red[0]);
}

__global__ void score_kernel(const float* __restrict__ x, const float* __restrict__ w,
                             const float* __restrict__ invnorm, float* score, int F) {
    __shared__ float red[256];
    int row = blockIdx.x;
    float s = 0.0f;
    for (int f = threadIdx.x; f < F; f += 256) s += x[(size_t)row * F + f] * w[f];
    red[threadIdx.x] = s;
    __syncthreads();
    for (int off = 128; off > 0; off >>= 1) {
        if (threadIdx.x < off) red[threadIdx.x] += red[threadIdx.x + off];
        __syncthreads();
    }
    if (threadIdx.x == 0) score[row] = tanhf(red[0] * invnorm[0]);
}

// Single-block bitonic sort top-k: descending score, ties -> lower index first
// (matches jax.lax.top_k). pow2 <= 4096.
__global__ __launch_bounds__(1024)
void topk_kernel(const float* __restrict__ score, int n, int k, int pow2,
                 int* perm, float* val) {
    __shared__ float s[4096];
    __shared__ int   id[4096];
    for (int i = threadIdx.x; i < pow2; i += blockDim.x) {
        if (i < n) { s[i] = score[i]; id[i] = i; }
        else       { s[i] = -__builtin_inff(); id[i] = 0x7fffffff; }
    }
    __syncthreads();
    for (int sz = 2; sz <= pow2; sz <<= 1) {
        for (int j = sz >> 1; j > 0; j >>= 1) {
            for (int i = threadIdx.x; i < pow2; i += blockDim.x) {
                int ixj = i ^ j;
                if (ixj > i) {
                    float si = s[i], sj = s[ixj];
                    int   ii = id[i], ij = id[ixj];
                    bool fwd = ((i & sz) == 0);
                    bool sw = fwd ? ((sj > si) || (sj == si && ij < ii))
                                  : ((si > sj) || (si == sj && ii < ij));
                    if (sw) { s[i] = sj; s[ixj] = si; id[i] = ij; id[ixj] = ii; }
                }
            }
            __syncthreads();
        }
    }
    for (int i = threadIdx.x; i < k; i += blockDim.x) { perm[i] = id[i]; val[i] = s[i]; }
}

__global__ void gather_x_kernel(const float* __restrict__ x, const int* __restrict__ perm,
                                const float* __restrict__ val, float* out, int total, int F) {
    int idx = blockIdx.x * 256 + threadIdx.x;
    if (idx < total) {
        int r = idx / F, f = idx % F;
        out[idx] = x[(size_t)perm[r] * F + f] * val[r];
    }
}

__global__ void gather_A_kernel(const float* __restrict__ Aaug, const int* __restrict__ perm,
                                float* out, int k, int nOld, int total) {
    int idx = blockIdx.x * 256 + threadIdx.x;
    if (idx < total) {
        int r = idx / k, c = idx % k;
        out[idx] = Aaug[(size_t)perm[r] * nOld + perm[c]];
    }
}

__global__ void copy_kernel(const float* __restrict__ src, float* dst, int total) {
    int i = blockIdx.x * 256 + threadIdx.x;
    if (i < total) dst[i] = src[i];
}

// perm rows are unique -> no atomics needed
__global__ void scatter_add_kernel(float* dst, const int* __restrict__ perm,
                                   const float* __restrict__ xlow, int total, int F) {
    int idx = blockIdx.x * 256 + threadIdx.x;
    if (idx < total) {
        int r = idx / F, f = idx % F;
        dst[(size_t)perm[r] * F + f] += xlow[idx];
    }
}

// ---------------------------------------------------------------------------
// Host orchestration
// ---------------------------------------------------------------------------
extern "C" void kernel_launch(void* const* d_in, const int* in_sizes, int n_in,
                              void* d_out, int out_size, void* d_ws, size_t ws_size,
                              hipStream_t stream) {
    const int Nn = 3000, F = 256, Cc = 40;
    const int KS0 = 2000, KS1 = 1000, KS2 = 500;

    const int*   ei      = (const int*)d_in[0];
    const float* x_param = (const float*)d_in[1];
    const float* W_down  = (const float*)d_in[2];   // (4,256,256)
    const float* b_down  = (const float*)d_in[3];   // (4,256)
    const float* pool_w  = (const float*)d_in[4];   // (3,256)
    const float* W_up    = (const float*)d_in[5];   // (2,256,256)
    const float* b_up    = (const float*)d_in[6];   // (2,256)
    const float* W_out   = (const float*)d_in[7];   // (40,256)
    const float* b_out   = (const float*)d_in[8];   // (40)
    const int E = in_sizes[0] / 2;                  // 96000 directed edges

    // ---- workspace layout -------------------------------------------------
    size_t off = 0;
    auto alloc = [&](size_t bytes) -> void* {
        void* p = (char*)d_ws + off;
        off += (bytes + 255) & ~(size_t)255;
        return p;
    };
    float* A0   = (float*)alloc((size_t)Nn * Nn * 4);
    float* A1b  = (float*)alloc((size_t)KS0 * KS0 * 4);
    float* A2b  = (float*)alloc((size_t)KS1 * KS1 * 4);
    float* A3b  = (float*)alloc((size_t)KS2 * KS2 * 4);
    float* Aaug = (float*)alloc((size_t)Nn * Nn * 4);
    float* A1t  = (float*)alloc((size_t)Nn * Nn * 4);
    float* XW   = (float*)alloc((size_t)Nn * F * 4);
    float* Zb   = (float*)alloc((size_t)Nn * F * 4);
    float* Yb   = (float*)alloc((size_t)Nn * F * 4);
    float* x0   = (float*)alloc((size_t)Nn * F * 4);
    float* x1   = (float*)alloc((size_t)KS0 * F * 4);
    float* x2   = (float*)alloc((size_t)KS1 * F * 4);
    float* t0   = (float*)alloc((size_t)Nn * F * 4);
    float* t1   = (float*)alloc((size_t)Nn * F * 4);
    float* dis  = (float*)alloc(Nn * 4);
    float* scr  = (float*)alloc(Nn * 4);
    float* val  = (float*)alloc(KS0 * 4);
    float* inrm = (float*)alloc(4);
    int*   perm0 = (int*)alloc(KS0 * 4);
    int*   perm1 = (int*)alloc(KS1 * 4);
    int*   perm2 = (int*)alloc(KS2 * 4);
    if (off > ws_size) return;   // refuse to scribble OOB

    auto cdiv = [](int a, int b) { return (a + b - 1) / b; };

    // TN selection: big tiles (best FLOP/LDS ratio) for the n^3 augment GEMMs,
    // medium for N=256 feature GEMMs (more blocks -> occupancy), small for N=40.
    auto gemm = [&](const float* A, const float* B, float* C, int M, int N, int K,
                    int lda, int ldb, int ldc, int transB) {
        if (N > 512) {
            dim3 grid(cdiv(N, 256), cdiv(M, 64));
            wmma_gemm_f32<256><<<grid, 256, 0, stream>>>(A, B, C, M, N, K, lda, ldb, ldc, transB);
        } else if (N > 64) {
            dim3 grid(cdiv(N, 128), cdiv(M, 64));
            wmma_gemm_f32<128><<<grid, 256, 0, stream>>>(A, B, C, M, N, K, lda, ldb, ldc, transB);
        } else {
            dim3 grid(cdiv(N, 64), cdiv(M, 64));
            wmma_gemm_f32<64><<<grid, 256, 0, stream>>>(A, B, C, M, N, K, lda, ldb, ldc, transB);
        }
    };

    // GCN: out = relu?( dis*(A@(dis*XW)) + 2*dis^2*XW + b ),  XW = x @ W^T
    auto gcn = [&](const float* A, int n, const float* xin, const float* W,
                   const float* bias, float* out, int Hout, int do_relu) {
        rowsum_dis_kernel<<<n, 256, 0, stream>>>(A, n, dis);
        gemm(xin, W, XW, n, Hout, F, F, F, Hout, 1);
        int tot = n * Hout;
        scale_rows_kernel<<<cdiv(tot, 256), 256, 0, stream>>>(XW, dis, Zb, tot, Hout);
        gemm(A, Zb, Yb, n, Hout, n, n, Hout, Hout, 0);
        gcn_epilogue_kernel<<<cdiv(tot, 256), 256, 0, stream>>>(Yb, XW, dis, bias, out,
                                                                tot, Hout, do_relu);
    };

    // augment: Aaug = (A with unit diag)^2, diag zeroed
    auto augment = [&](const float* A, int n) {
        int tot = n * n;
        make_a1_kernel<<<cdiv(tot, 256), 256, 0, stream>>>(A, A1t, n, tot);
        gemm(A1t, A1t, Aaug, n, n, n, n, n, n, 0);
        zero_diag_kernel<<<cdiv(n, 256), 256, 0, stream>>>(Aaug, n);
    };

    // TopK pool: scores from xin, select k, gather x (scaled) and Aaug submatrix
    auto pool = [&](const float* xin, int n, int k, int pow2, const float* w,
                    int* perm, float* xout, float* Aout) {
        invnorm_kernel<<<1, 256, 0, stream>>>(w, F, inrm);
        score_kernel<<<n, 256, 0, stream>>>(xin, w, inrm, scr, F);
        topk_kernel<<<1, 1024, 0, stream>>>(scr, n, k, pow2, perm, val);
        int tx = k * F;
        gather_x_kernel<<<cdiv(tx, 256), 256, 0, stream>>>(xin, perm, val, xout, tx, F);
        int tA = k * k;
        gather_A_kernel<<<cdiv(tA, 256), 256, 0, stream>>>(Aaug, perm, Aout, k, n, tA);
    };

    // ---- build adjacency --------------------------------------------------
    int totA0 = Nn * Nn;
    fill_zero_kernel<<<cdiv(totA0, 256), 256, 0, stream>>>(A0, totA0);
    build_adj_kernel<<<cdiv(E, 256), 256, 0, stream>>>(ei, E, A0, Nn);

    // ---- down path --------------------------------------------------------
    gcn(A0, Nn, x_param, W_down, b_down, x0, F, 1);

    augment(A0, Nn);
    pool(x0, Nn, KS0, 4096, pool_w, perm0, t0, A1b);
    gcn(A1b, KS0, t0, W_down + 1 * F * F, b_down + 1 * F, x1, F, 1);

    augment(A1b, KS0);
    pool(x1, KS0, KS1, 2048, pool_w + 1 * F, perm1, t0, A2b);
    gcn(A2b, KS1, t0, W_down + 2 * F * F, b_down + 2 * F, x2, F, 1);

    augment(A2b, KS1);
    pool(x2, KS1, KS2, 1024, pool_w + 2 * F, perm2, t0, A3b);
    gcn(A3b, KS2, t0, W_down + 3 * F * F, b_down + 3 * F, t1, F, 1);  // deepest x in t1

    // ---- up path ----------------------------------------------------------
    // j = 2 : size KS1, skip x2, scatter t1 (KS2 rows) via perm2
    copy_kernel<<<cdiv(KS1 * F, 256), 256, 0, stream>>>(x2, t0, KS1 * F);
    scatter_add_kernel<<<cdiv(KS2 * F, 256), 256, 0, stream>>>(t0, perm2, t1, KS2 * F, F);
    gcn(A2b, KS1, t0, W_up, b_up, t1, F, 1);

    // j = 1 : size KS0, skip x1, scatter t1 (KS1 rows) via perm1
    copy_kernel<<<cdiv(KS0 * F, 256), 256, 0, stream>>>(x1, t0, KS0 * F);
    scatter_add_kernel<<<cdiv(KS1 * F, 256), 256, 0, stream>>>(t0, perm1, t1, KS1 * F, F);
    gcn(A1b, KS0, t0, W_up + 1 * F * F, b_up + 1 * F, t1, F, 1);

    // j = 0 : size N, skip x0, scatter t1 (KS0 rows) via perm0, final GCN -> d_out
    copy_kernel<<<cdiv(Nn * F, 256), 256, 0, stream>>>(x0, t0, Nn * F);
    scatter_add_kernel<<<cdiv(KS0 * F, 256), 256, 0, stream>>>(t0, perm0, t1, KS0 * F, F);
    gcn(A0, Nn, t0, W_out, b_out, (float*)d_out, Cc, 0);
}